// Gat_45114336477374
// MI455X (gfx1250) — compile-verified
//
#include <hip/hip_runtime.h>

#define NN   512
#define FIN  256
#define HF   64
#define KTOP 12
#define WH_STRIDE 65   // 64 + 1 pad to break LDS bank alignment

typedef __attribute__((ext_vector_type(2))) float v2f;
typedef __attribute__((ext_vector_type(8))) float v8f;

// D = A(16x4, f32) * B(4x16, f32) + C(16x16, f32)
__device__ __forceinline__ v8f wmma4f32(v2f a, v2f b, v8f c) {
    return __builtin_amdgcn_wmma_f32_16x16x4_f32(
        /*neg_a=*/false, a, /*neg_b=*/false, b,
        /*c_mod=*/(short)0, c, /*reuse_a=*/false, /*reuse_b=*/false);
}

__device__ __forceinline__ float lrelu02(float x) {
    // leaky_relu with slope 0.2: for x<0, 0.2x > x, so max() implements it
    return fmaxf(x, 0.2f * x);
}
__device__ __forceinline__ float elu1(float x) {
    return (x > 0.0f) ? x : (__expf(x) - 1.0f);
}

// ---------------------------------------------------------------------------
// Kernel A: Avv = V @ V^T   (512x64 @ 64x512)
// ---------------------------------------------------------------------------
__global__ __launch_bounds__(256) void vv_kernel(const float* __restrict__ V,
                                                 float* __restrict__ Avv) {
    int idx = blockIdx.x * 256 + threadIdx.x;     // 0..262143
    int i = idx >> 9, j = idx & 511;
    const float* vi = V + (size_t)i * HF;
    const float* vj = V + (size_t)j * HF;
    float s = 0.0f;
#pragma unroll 8
    for (int k = 0; k < HF; ++k) s = fmaf(vi[k], vj[k], s);
    Avv[idx] = s;
}

// ---------------------------------------------------------------------------
// Kernel B: kth = 12th largest of Avv; adj = (Avv > kth) ? 1 : 0
// Writes adj to workspace AND to d_out (first returned tensor).
// ---------------------------------------------------------------------------
__global__ __launch_bounds__(1024) void topk_kernel(const float* __restrict__ Avv,
                                                    float* __restrict__ adj_ws,
                                                    float* __restrict__ adj_out) {
    __shared__ float cand[1024 * KTOP];
    __shared__ float s_kth;
    const int tid = threadIdx.x;

    float loc[KTOP];
#pragma unroll
    for (int t = 0; t < KTOP; ++t) loc[t] = -3.402823466e38f;

    for (int idx = tid; idx < NN * NN; idx += 1024) {
        float v = Avv[idx];
        if (v > loc[KTOP - 1]) {
            int p = KTOP - 1;
            while (p > 0 && loc[p - 1] < v) { loc[p] = loc[p - 1]; --p; }
            loc[p] = v;
        }
    }
#pragma unroll
    for (int t = 0; t < KTOP; ++t) cand[tid * KTOP + t] = loc[t];
    __syncthreads();

    if (tid == 0) {
        float g[KTOP];
#pragma unroll
        for (int t = 0; t < KTOP; ++t) g[t] = -3.402823466e38f;
        for (int i = 0; i < 1024 * KTOP; ++i) {
            float v = cand[i];
            if (v > g[KTOP - 1]) {
                int p = KTOP - 1;
                while (p > 0 && g[p - 1] < v) { g[p] = g[p - 1]; --p; }
                g[p] = v;
            }
        }
        s_kth = g[KTOP - 1];
    }
    __syncthreads();

    const float kth = s_kth;
    for (int idx = tid; idx < NN * NN; idx += 1024) {
        float m = (Avv[idx] > kth) ? 1.0f : 0.0f;
        adj_ws[idx]  = m;
        adj_out[idx] = m;
    }
}

// ---------------------------------------------------------------------------
// Kernel C: per-head GAT layer (grid.x = 512 heads, 256 threads = 8 waves).
//   Wh = X @ W_h            (WMMA f32 16x16x4, Wh kept in LDS)
//   e1/e2, softmax stats    (VALU + __expf)
//   out = mask(softmax(S)) @ Wh  (WMMA, A-frags built on the fly)
//   mid[:, h*64 : h*64+64] = elu(out)
// ---------------------------------------------------------------------------
__global__ __launch_bounds__(256) void heads_kernel(
    const float* __restrict__ X,        // (512,256)
    const float* __restrict__ W_stack,  // (512,256,64)
    const float* __restrict__ a_stack,  // (512,128)
    const float* __restrict__ adj,      // (512,512) in ws (symmetric)
    float* __restrict__ mid)            // (512, 32768)
{
    __shared__ float sWh[NN * WH_STRIDE];
    __shared__ float sE1[NN], sE2[NN], sMax[NN], sInv[NN];
    __shared__ float sMxE2;

    const int head = blockIdx.x;
    const int tid  = threadIdx.x;
    const int lane = tid & 31;
    const int wave = tid >> 5;
    const int half = lane >> 4;     // 0: lanes 0-15, 1: lanes 16-31
    const int l16  = lane & 15;

    const float* W  = W_stack + (size_t)head * FIN * HF;
    const float* a1 = a_stack + (size_t)head * 2 * HF;
    const float* a2 = a1 + HF;

    // ---- Phase 1: Wh = X @ W  ------------------------------------------
#pragma unroll 1
    for (int mt = wave * 4; mt < wave * 4 + 4; ++mt) {
        const int m0  = mt * 16;
        const int row = m0 + l16;
        const float* xr = X + (size_t)row * FIN;
        v8f acc[4] = {{}, {}, {}, {}};
        for (int k0 = 0; k0 < FIN; k0 += 4) {
            const int ka = k0 + 2 * half;
            v2f af;
            af.x = xr[ka];
            af.y = xr[ka + 1];
#pragma unroll
            for (int nt = 0; nt < 4; ++nt) {
                v2f bf;
                bf.x = W[(size_t)ka * HF + nt * 16 + l16];
                bf.y = W[(size_t)(ka + 1) * HF + nt * 16 + l16];
                acc[nt] = wmma4f32(af, bf, acc[nt]);
            }
        }
        // C/D layout: VGPR r, lane L -> M = r + 8*half, N = l16
#pragma unroll
        for (int nt = 0; nt < 4; ++nt)
#pragma unroll
            for (int r = 0; r < 8; ++r)
                sWh[(m0 + r + 8 * half) * WH_STRIDE + nt * 16 + l16] = acc[nt][r];
    }
    __syncthreads();

    // ---- Phase 2: e1, e2, row max, softmax denominators -----------------
    for (int i = tid; i < NN; i += 256) {
        float s1 = 0.0f, s2 = 0.0f;
        const float* wr = &sWh[i * WH_STRIDE];
#pragma unroll 8
        for (int c = 0; c < HF; ++c) {
            float w = wr[c];
            s1 = fmaf(w, a1[c], s1);
            s2 = fmaf(w, a2[c], s2);
        }
        sE1[i] = s1;
        sE2[i] = s2;
    }
    __syncthreads();
    if (tid == 0) {
        float mx = sE2[0];
        for (int j = 1; j < NN; ++j) mx = fmaxf(mx, sE2[j]);
        sMxE2 = mx;
    }
    __syncthreads();
    for (int i = tid; i < NN; i += 256) {
        float e1 = sE1[i];
        float mx = lrelu02(e1 + sMxE2);   // LReLU monotone -> row max
        float s  = 0.0f;
        for (int j = 0; j < NN; ++j)
            s += __expf(lrelu02(e1 + sE2[j]) - mx);
        sMax[i] = mx;
        sInv[i] = 1.0f / s;
    }
    __syncthreads();

    // ---- Phase 3: out = A_masked @ Wh, elu, write to mid ----------------
#pragma unroll 1
    for (int mt = wave * 4; mt < wave * 4 + 4; ++mt) {
        const int m0 = mt * 16;
        const int m  = m0 + l16;       // A-frag row per lane (both halves)
        const float e1m = sE1[m];
        const float mxm = sMax[m];
        const float ivm = sInv[m];
        v8f acc[4] = {{}, {}, {}, {}};
        for (int k0 = 0; k0 < NN; k0 += 4) {
            const int j0 = k0 + 2 * half;
            // adj is symmetric -> read adj[j][m] for coalesced lanes
            float msk0 = adj[(size_t)j0 * NN + m];
            float msk1 = adj[(size_t)(j0 + 1) * NN + m];
            float ex0 = __expf(lrelu02(e1m + sE2[j0])     - mxm) * ivm;
            float ex1 = __expf(lrelu02(e1m + sE2[j0 + 1]) - mxm) * ivm;
            v2f af;
            af.x = (msk0 > 0.0f) ? ex0 : 0.0f;
            af.y = (msk1 > 0.0f) ? ex1 : 0.0f;
#pragma unroll
            for (int nt = 0; nt < 4; ++nt) {
                v2f bf;
                bf.x = sWh[j0 * WH_STRIDE + nt * 16 + l16];
                bf.y = sWh[(j0 + 1) * WH_STRIDE + nt * 16 + l16];
                acc[nt] = wmma4f32(af, bf, acc[nt]);
            }
        }
#pragma unroll
        for (int nt = 0; nt < 4; ++nt)
#pragma unroll
            for (int r = 0; r < 8; ++r) {
                int row = m0 + r + 8 * half;
                int col = nt * 16 + l16;
                mid[(size_t)row * (NN * HF) + (size_t)head * HF + col] =
                    elu1(acc[nt][r]);
            }
    }
}

// ---------------------------------------------------------------------------
// Kernel D: Wh2 = mid (512x32768) @ W_out (32768x64)
// grid = 32 m-tiles * 4 n-tiles = 128 blocks, 8 waves split K, LDS reduce.
// ---------------------------------------------------------------------------
__global__ __launch_bounds__(256) void outproj_kernel(
    const float* __restrict__ mid,
    const float* __restrict__ W_out,
    float* __restrict__ Wh2)
{
    __shared__ float red[8 * 256];
    const int mt = blockIdx.x >> 2;
    const int nt = blockIdx.x & 3;
    const int m0 = mt * 16, n0 = nt * 16;
    const int tid = threadIdx.x, lane = tid & 31, wave = tid >> 5;
    const int half = lane >> 4, l16 = lane & 15;
    const int Ktot = NN * HF;                 // 32768
    const int kbeg = wave * (Ktot / 8);
    const int kend = kbeg + (Ktot / 8);
    const int row  = m0 + l16;
    const float* mr = mid + (size_t)row * Ktot;

    v8f acc = {};
    for (int k0 = kbeg; k0 < kend; k0 += 4) {
        const int ka = k0 + 2 * half;
        v2f af;
        af.x = mr[ka];
        af.y = mr[ka + 1];
        v2f bf;
        bf.x = W_out[(size_t)ka * HF + n0 + l16];
        bf.y = W_out[(size_t)(ka + 1) * HF + n0 + l16];
        acc = wmma4f32(af, bf, acc);
    }
#pragma unroll
    for (int r = 0; r < 8; ++r)
        red[wave * 256 + (r + 8 * half) * 16 + l16] = acc[r];
    __syncthreads();

    float s = 0.0f;
#pragma unroll
    for (int w = 0; w < 8; ++w) s += red[w * 256 + tid];
    const int r = tid >> 4, c = tid & 15;
    Wh2[(size_t)(m0 + r) * HF + n0 + c] = s;
}

// ---------------------------------------------------------------------------
// Kernel E: output attention layer + elu(V * elu(out)), single workgroup.
// ---------------------------------------------------------------------------
__global__ __launch_bounds__(256) void final_kernel(
    const float* __restrict__ Wh2,     // (512,64) in ws
    const float* __restrict__ a_out,   // (128)
    const float* __restrict__ adj,     // (512,512) in ws
    const float* __restrict__ V,       // (512,64)
    float* __restrict__ out)           // d_out + 512*512
{
    __shared__ float sWh[NN * WH_STRIDE];
    __shared__ float sE1[NN], sE2[NN], sMax[NN], sInv[NN];
    __shared__ float sMxE2;

    const int tid  = threadIdx.x;
    const int lane = tid & 31;
    const int wave = tid >> 5;
    const int half = lane >> 4;
    const int l16  = lane & 15;
    const float* a1 = a_out;
    const float* a2 = a_out + HF;

    for (int i = tid; i < NN * HF; i += 256)
        sWh[(i / HF) * WH_STRIDE + (i % HF)] = Wh2[i];
    __syncthreads();

    for (int i = tid; i < NN; i += 256) {
        float s1 = 0.0f, s2 = 0.0f;
        const float* wr = &sWh[i * WH_STRIDE];
#pragma unroll 8
        for (int c = 0; c < HF; ++c) {
            float w = wr[c];
            s1 = fmaf(w, a1[c], s1);
            s2 = fmaf(w, a2[c], s2);
        }
        sE1[i] = s1;
        sE2[i] = s2;
    }
    __syncthreads();
    if (tid == 0) {
        float mx = sE2[0];
        for (int j = 1; j < NN; ++j) mx = fmaxf(mx, sE2[j]);
        sMxE2 = mx;
    }
    __syncthreads();
    for (int i = tid; i < NN; i += 256) {
        float e1 = sE1[i];
        float mx = lrelu02(e1 + sMxE2);
        float s  = 0.0f;
        for (int j = 0; j < NN; ++j)
            s += __expf(lrelu02(e1 + sE2[j]) - mx);
        sMax[i] = mx;
        sInv[i] = 1.0f / s;
    }
    __syncthreads();

#pragma unroll 1
    for (int mt = wave * 4; mt < wave * 4 + 4; ++mt) {
        const int m0 = mt * 16;
        const int m  = m0 + l16;
        const float e1m = sE1[m];
        const float mxm = sMax[m];
        const float ivm = sInv[m];
        v8f acc[4] = {{}, {}, {}, {}};
        for (int k0 = 0; k0 < NN; k0 += 4) {
            const int j0 = k0 + 2 * half;
            float msk0 = adj[(size_t)j0 * NN + m];
            float msk1 = adj[(size_t)(j0 + 1) * NN + m];
            float ex0 = __expf(lrelu02(e1m + sE2[j0])     - mxm) * ivm;
            float ex1 = __expf(lrelu02(e1m + sE2[j0 + 1]) - mxm) * ivm;
            v2f af;
            af.x = (msk0 > 0.0f) ? ex0 : 0.0f;
            af.y = (msk1 > 0.0f) ? ex1 : 0.0f;
#pragma unroll
            for (int nt = 0; nt < 4; ++nt) {
                v2f bf;
                bf.x = sWh[j0 * WH_STRIDE + nt * 16 + l16];
                bf.y = sWh[(j0 + 1) * WH_STRIDE + nt * 16 + l16];
                acc[nt] = wmma4f32(af, bf, acc[nt]);
            }
        }
#pragma unroll
        for (int nt = 0; nt < 4; ++nt)
#pragma unroll
            for (int r = 0; r < 8; ++r) {
                int row = m0 + r + 8 * half;
                int col = nt * 16 + l16;
                float t = elu1(acc[nt][r]);               // gat_layer elu
                float y = V[(size_t)row * HF + col] * t;  // V * out
                out[(size_t)row * HF + col] = elu1(y);    // final elu
            }
    }
}

// ---------------------------------------------------------------------------
// Host-side launcher
// ---------------------------------------------------------------------------
extern "C" void kernel_launch(void* const* d_in, const int* in_sizes, int n_in,
                              void* d_out, int out_size, void* d_ws, size_t ws_size,
                              hipStream_t stream) {
    const float* X       = (const float*)d_in[0];
    const float* W_stack = (const float*)d_in[1];
    const float* a_stack = (const float*)d_in[2];
    const float* W_out   = (const float*)d_in[3];
    const float* a_out   = (const float*)d_in[4];
    const float* V       = (const float*)d_in[5];

    float* out = (float*)d_out;  // [0 .. 512*512) = adj, then (512,64) final

    // workspace layout (floats)
    float* ws   = (float*)d_ws;
    float* Avv  = ws;                                   // 512*512
    float* adjw = ws + (size_t)NN * NN;                 // 512*512
    float* mid  = ws + (size_t)2 * NN * NN;             // 512*32768
    float* Wh2  = mid + (size_t)NN * NN * HF;           // 512*64

    vv_kernel     <<<(NN * NN) / 256, 256, 0, stream>>>(V, Avv);
    topk_kernel   <<<1, 1024, 0, stream>>>(Avv, adjw, out);
    heads_kernel  <<<NN, 256, 0, stream>>>(X, W_stack, a_stack, adjw, mid);
    outproj_kernel<<<128, 256, 0, stream>>>(mid, W_out, Wh2);
    final_kernel  <<<1, 256, 0, stream>>>(Wh2, a_out, adjw, V,
                                          out + (size_t)NN * NN);
}